// GNNComplete_52312701665407
// MI455X (gfx1250) — compile-verified
//
#include <hip/hip_runtime.h>
#include <hip/hip_bf16.h>

// CDNA5 / gfx1250 GCN forward.
//   layer: y = (x @ W) * dis[row]      (fp32 WMMA 16x16x4, scale fused in epilogue)
//          h = relu(dis[c]*(y[c] + sum_{e: col=c} y[row_e]) + b)   (CSC gather-reduce)
//   out  : h2 @ Wm + bm                (fp32 WMMA, bias fused)
// A tiles staged to LDS via GLOBAL_LOAD_ASYNC_TO_LDS_B128 (ASYNCcnt).
// Workspace: ~213 MB (CSC arrays + two 100k x 256 fp32 feature buffers).

typedef __attribute__((ext_vector_type(2))) float v2f;
typedef __attribute__((ext_vector_type(8))) float v8f;

__global__ __launch_bounds__(256) void k_zero_i32(int* __restrict__ p, int n) {
  int i = blockIdx.x * blockDim.x + threadIdx.x;
  if (i < n) p[i] = 0;
}

__global__ __launch_bounds__(256) void k_count(const int* __restrict__ col, int E,
                                               int* __restrict__ cnt) {
  int i = blockIdx.x * blockDim.x + threadIdx.x;
  if (i < E) atomicAdd(&cnt[col[i]], 1);
}

// Single-workgroup exclusive scan over node degrees; also emits dis = rsqrt(deg+1)
// (the +1 is the self-loop) and initializes the bucket cursors.
__global__ __launch_bounds__(1024) void k_scan(const int* __restrict__ cnt,
                                               int* __restrict__ offs,
                                               int* __restrict__ cursor,
                                               float* __restrict__ dis, int n) {
  __shared__ int sdat[1024];
  __shared__ int carry;
  if (threadIdx.x == 0) carry = 0;
  __syncthreads();
  for (int base = 0; base < n; base += 1024) {
    int i = base + (int)threadIdx.x;
    int v = (i < n) ? cnt[i] : 0;
    sdat[threadIdx.x] = v;
    __syncthreads();
    for (int off = 1; off < 1024; off <<= 1) {
      int t = (threadIdx.x >= (unsigned)off) ? sdat[threadIdx.x - off] : 0;
      __syncthreads();
      sdat[threadIdx.x] += t;
      __syncthreads();
    }
    int incl = sdat[threadIdx.x];
    int excl = incl - v;
    if (i < n) {
      int o = carry + excl;
      offs[i]   = o;
      cursor[i] = o;
      dis[i]    = rsqrtf((float)(v + 1));   // deg >= 1 always (self-loop)
    }
    __syncthreads();
    if (threadIdx.x == 1023) carry += sdat[1023];
    __syncthreads();
  }
  if (threadIdx.x == 0) offs[n] = carry;
}

__global__ __launch_bounds__(256) void k_bucket(const int* __restrict__ row,
                                                const int* __restrict__ col, int E,
                                                int* __restrict__ cursor,
                                                int* __restrict__ src) {
  int i = blockIdx.x * blockDim.x + threadIdx.x;
  if (i < E) {
    int pos = atomicAdd(&cursor[col[i]], 1);
    src[pos] = row[i];
  }
}

// Canonicalize each bucket (ascending) so the float reduction order is
// deterministic across calls despite the atomic bucketing above.
__global__ __launch_bounds__(256) void k_sort(const int* __restrict__ offs,
                                              int* __restrict__ src, int n) {
  int c = blockIdx.x * blockDim.x + threadIdx.x;
  if (c >= n) return;
  int s = offs[c], e = offs[c + 1];
  for (int i = s + 1; i < e; ++i) {
    int v = src[i];
    int j = i - 1;
    while (j >= s && src[j] > v) { src[j + 1] = src[j]; --j; }
    src[j + 1] = v;
  }
}

// C = A[Mx256] @ B[256xN], fp32 WMMA 16x16x4.
// Block = 256 threads (8 waves). Each wave owns NT independent 16x16 N-tiles
// (cols n0 and n0+128) sharing one A fragment -> 2 interleaved WMMA chains.
// NT=2,N=256: block covers 16x256. NT=1,N=128: block covers 16x128. Grid = M/16.
template <int NT, bool SCALE, bool BIAS>
__global__ __launch_bounds__(256) void k_gemm_wmma(const float* __restrict__ A,
                                                   const float* __restrict__ B,
                                                   const float* __restrict__ dis,
                                                   const float* __restrict__ bias,
                                                   float* __restrict__ Y, int N) {
  __shared__ float As[16][260];           // pitch 260: conflict-free b64 reads
  const int row0 = blockIdx.x * 16;

  // Async-stage A tile (16 rows x 256 K) into LDS: 4 x b128 per thread,
  // tracked by ASYNCcnt (CDNA5 ISA).
  {
    const float* gsrc = A + (size_t)row0 * 256;
#pragma unroll
    for (int t = 0; t < 4; ++t) {
      int e4 = (int)threadIdx.x + t * 256;   // float4 index, 0..1023
      int m  = e4 >> 6;
      int k4 = e4 & 63;
      unsigned ldsAddr = (unsigned)(uintptr_t)&As[m][k4 * 4]; // LDS aperture: addr[31:0]
      const float* g = gsrc + (size_t)m * 256 + (size_t)k4 * 4;
      asm volatile("global_load_async_to_lds_b128 %0, %1, off"
                   :: "v"(ldsAddr), "v"(g) : "memory");
    }
    asm volatile("s_wait_asynccnt 0" ::: "memory");
  }
  __syncthreads();

  const int lane = (int)threadIdx.x & 31;
  const int wave = (int)threadIdx.x >> 5;
  const int hi   = lane >> 4;              // half-wave select
  const int mSel = lane & 15;
  const int n0   = wave * 16 + (lane & 15);

  v8f c[NT];
  const v8f vzero = {};
#pragma unroll
  for (int t = 0; t < NT; ++t) c[t] = vzero;

  for (int k0 = 0; k0 < 256; k0 += 4) {
    const int kb = k0 + 2 * hi;            // ISA f32 A/B frag: lanes16-31 hold K+2,K+3
    float2 av = *(const float2*)&As[mSel][kb];
    v2f a; a.x = av.x; a.y = av.y;
#pragma unroll
    for (int t = 0; t < NT; ++t) {
      const int n = n0 + t * 128;
      v2f b;
      b.x = B[(size_t)kb * N + n];
      b.y = B[(size_t)(kb + 1) * N + n];
      c[t] = __builtin_amdgcn_wmma_f32_16x16x4_f32(false, a, false, b, (short)0,
                                                   c[t], false, false);
    }
  }

  // D layout: VGPR r -> M = r + 8*hi, N = lane&15.
  float bs[NT];
#pragma unroll
  for (int t = 0; t < NT; ++t) bs[t] = BIAS ? bias[n0 + t * 128] : 0.0f;

#pragma unroll
  for (int r = 0; r < 8; ++r) {
    const int grow = row0 + r + 8 * hi;
    const float sc = SCALE ? dis[grow] : 1.0f;
#pragma unroll
    for (int t = 0; t < NT; ++t) {
      float v = c[t][r];
      if (SCALE) v *= sc;
      if (BIAS)  v += bs[t];
      Y[(size_t)grow * N + (n0 + t * 128)] = v;
    }
  }
}

// h[c,:] = relu(dis[c] * (y[c,:] + sum_{p in [offs[c],offs[c+1])} y[src[p],:]) + b)
// One 256-thread block per node; edge indices staged through LDS; the gather
// source (102 MB) is L2-resident on the 192 MB L2.
__global__ __launch_bounds__(256) void k_aggregate(const float* __restrict__ Y,
                                                   const int* __restrict__ offs,
                                                   const int* __restrict__ src,
                                                   const float* __restrict__ dis,
                                                   const float* __restrict__ bias,
                                                   float* __restrict__ H) {
  __shared__ int sSrc[256];
  const int c = blockIdx.x;
  const int j = threadIdx.x;
  float acc = Y[(size_t)c * 256 + j];       // self-loop term
  const int s = offs[c], e = offs[c + 1];
  for (int base = s; base < e; base += 256) {
    int chunk = min(256, e - base);
    __syncthreads();
    if (j < chunk) sSrc[j] = src[base + j];
    __syncthreads();
    for (int t = 0; t < chunk; ++t)
      acc += Y[(size_t)sSrc[t] * 256 + j];
  }
  H[(size_t)c * 256 + j] = fmaxf(fmaf(acc, dis[c], bias[j]), 0.0f);
}

extern "C" void kernel_launch(void* const* d_in, const int* in_sizes, int n_in,
                              void* d_out, int out_size, void* d_ws, size_t ws_size,
                              hipStream_t stream) {
  const float* x  = (const float*)d_in[0];
  const int*   ei = (const int*)d_in[1];
  const float* W1 = (const float*)d_in[2];
  const float* b1 = (const float*)d_in[3];
  const float* W2 = (const float*)d_in[4];
  const float* b2 = (const float*)d_in[5];
  const float* Wm = (const float*)d_in[6];
  const float* bm = (const float*)d_in[7];
  float* out = (float*)d_out;

  const int Nn  = in_sizes[0] / 256;   // 100000
  const int E   = in_sizes[1] / 2;     // 1600000
  const int OUT = out_size / Nn;       // 128
  (void)OUT;

  const int* rowp = ei;        // edge_index[0]
  const int* colp = ei + E;    // edge_index[1]

  char* ws = (char*)d_ws;
  auto carve = [&](size_t bytes) {
    void* p = (void*)ws;
    ws += (bytes + 255) & ~(size_t)255;
    return p;
  };
  int*   cnt    = (int*)  carve(sizeof(int)   * (size_t)Nn);
  int*   offs   = (int*)  carve(sizeof(int)   * (size_t)(Nn + 1));
  int*   cursor = (int*)  carve(sizeof(int)   * (size_t)Nn);
  float* dis    = (float*)carve(sizeof(float) * (size_t)Nn);
  int*   src    = (int*)  carve(sizeof(int)   * (size_t)E);
  float* y      = (float*)carve(sizeof(float) * (size_t)Nn * 256);
  float* h      = (float*)carve(sizeof(float) * (size_t)Nn * 256);

  // ---- CSC build (deterministic after per-bucket sort) ----
  k_zero_i32<<<(Nn + 255) / 256, 256, 0, stream>>>(cnt, Nn);
  k_count   <<<(E  + 255) / 256, 256, 0, stream>>>(colp, E, cnt);
  k_scan    <<<1, 1024, 0, stream>>>(cnt, offs, cursor, dis, Nn);
  k_bucket  <<<(E  + 255) / 256, 256, 0, stream>>>(rowp, colp, E, cursor, src);
  k_sort    <<<(Nn + 255) / 256, 256, 0, stream>>>(offs, src, Nn);

  // ---- layer 1 ----
  k_gemm_wmma<2, true, false><<<Nn / 16, 256, 0, stream>>>(x, W1, dis, nullptr, y, 256);
  k_aggregate<<<Nn, 256, 0, stream>>>(y, offs, src, dis, b1, h);

  // ---- layer 2 ----
  k_gemm_wmma<2, true, false><<<Nn / 16, 256, 0, stream>>>(h, W2, dis, nullptr, y, 256);
  k_aggregate<<<Nn, 256, 0, stream>>>(y, offs, src, dis, b2, h);

  // ---- output projection (N = 128, single tile per wave) ----
  k_gemm_wmma<1, false, true><<<Nn / 16, 256, 0, stream>>>(h, Wm, nullptr, bm, out, 128);
}